// GripperRegionNetwork_16381005267637
// MI455X (gfx1250) — compile-verified
//
#include <hip/hip_runtime.h>
#include <hip/hip_bf16.h>

#define B_      512
#define G_      8192
#define REGION  1024
#define THREADS 256
#define WAVES   8        // THREADS / 32
#define STRIP   1024     // G_ / WAVES : points per wave
#define PTS     16       // points per WMMA
#define NCHUNK  (STRIP / PTS)

typedef __attribute__((ext_vector_type(2))) float v2f;
typedef __attribute__((ext_vector_type(8))) float v8f;

__global__ __launch_bounds__(THREADS)
void gripper_region_kernel(const float* __restrict__ gp,      // (B,G,6)
                           const int*   __restrict__ gindex,  // (B,G)
                           const float* __restrict__ grasp,   // (B,7)
                           const float* __restrict__ widths_p,
                           const float* __restrict__ height_p,
                           const float* __restrict__ depths_p,
                           float* __restrict__ out_pc,        // (B,REGION,6)
                           float* __restrict__ out_idx,       // (B,REGION)
                           float* __restrict__ out_inall,     // (B,REGION)
                           float* __restrict__ out_mask)      // (B,)
{
    __shared__ float s_mat[12];               // Mrows row-major (9) + center (3)
    __shared__ int   s_list[WAVES * STRIP];   // per-wave ordered index segments
    __shared__ int   s_cnt[WAVES];

    const int b    = blockIdx.x;
    const int tid  = threadIdx.x;
    const int lane = tid & 31;
    const int wav  = tid >> 5;

    const float ylim = 0.5f * widths_p[0];
    const float zlim = 0.5f * height_p[0];
    const float xlim = 0.5f * depths_p[0];

    // ---- build frame (one thread, mirrors _transform) ----
    if (tid == 0) {
        const float* gr = grasp + b * 7;
        float cx = gr[0], cy = gr[1], cz = gr[2];
        float ayx = gr[3], ayy = gr[4], ayz = gr[5];
        float ang = gr[6];
        float cth = cosf(ang), sth = sinf(ang);
        // axis_y = normalize_fix(axis_y, [0,1,0])
        float nn  = sqrtf(ayx*ayx + ayy*ayy + ayz*ayz);
        float inv = 1.0f / (nn + 1e-12f);
        float yx = ayx*inv, yy = ayy*inv, yz = ayz*inv;
        if (nn == 0.0f) { yx = 0.f; yy = 1.f; yz = 0.f; }
        // axis_x = normalize_fix([y.y, -y.x, 0], [1,0,0])
        float txx = yy, txy = -yx;
        nn  = sqrtf(txx*txx + txy*txy);
        inv = 1.0f / (nn + 1e-12f);
        float xx = txx*inv, xy = txy*inv, xz = 0.f;
        if (nn == 0.0f) { xx = 1.f; xy = 0.f; xz = 0.f; }
        // axis_z = cross(axis_x, axis_y), fixed-normalize to [0,0,1]
        float zx = xy*yz - xz*yy;
        float zy = xz*yx - xx*yz;
        float zz = xx*yy - xy*yx;
        float nz = sqrtf(zx*zx + zy*zy + zz*zz);
        float dv = (nz == 0.f) ? 1.f : nz;
        zx /= dv; zy /= dv; zz /= dv;
        if (nz == 0.f) { zx = 0.f; zy = 0.f; zz = 1.f; }
        // approach = normalize_fix( (S @ R1)[:,0] ) = normalize(ax*c + az*s)
        float apx = xx*cth + zx*sth;
        float apy = xy*cth + zy*sth;
        float apz = xz*cth + zz*sth;
        nn  = sqrtf(apx*apx + apy*apy + apz*apz);
        inv = 1.0f / (nn + 1e-12f);
        apx *= inv; apy *= inv; apz *= inv;
        if (nn == 0.f) { apx = 1.f; apy = 0.f; apz = 0.f; }
        // minor = cross(approach, axis_y)
        float mx = apy*yz - apz*yy;
        float my = apz*yx - apx*yz;
        float mz = apx*yy - apy*yx;
        s_mat[0] = apx; s_mat[1] = apy; s_mat[2] = apz;   // row 0
        s_mat[3] = yx;  s_mat[4] = yy;  s_mat[5] = yz;    // row 1
        s_mat[6] = mx;  s_mat[7] = my;  s_mat[8] = mz;    // row 2
        s_mat[9] = cx;  s_mat[10] = cy; s_mat[11] = cz;
    }
    __syncthreads();

    // ---- per-lane WMMA B and C operands ----
    // B[K][N] = Mrows[N][K] (first 3 cols), layout mirrors A:
    //   VGPR0: lanes 0-15 -> K=0, lanes 16-31 -> K=2
    //   VGPR1: lanes 0-15 -> K=1, lanes 16-31 -> K=3 (zero pad)
    const int ncol = lane & 15;
    v2f bmat; bmat.x = 0.f; bmat.y = 0.f;
    float cval = 0.f;
    if (ncol < 3) {
        float r0 = s_mat[ncol*3 + 0];
        float r1 = s_mat[ncol*3 + 1];
        float r2 = s_mat[ncol*3 + 2];
        cval = -(r0*s_mat[9] + r1*s_mat[10] + r2*s_mat[11]);
        if (lane < 16) { bmat.x = r0; bmat.y = r1; }
        else           { bmat.x = r2; bmat.y = 0.f; }
    }
    v8f cmat;
#pragma unroll
    for (int i = 0; i < 8; ++i) cmat[i] = cval;

    // per-lane bounds for the D column this lane holds (cols >= 3 are don't-care)
    const float blo = (ncol == 0) ? 0.f  : ((ncol == 1) ? -ylim : -zlim);
    const float bhi = (ncol == 0) ? xlim : ((ncol == 1) ?  ylim :  zlim);

    // ---- barrier-free streaming scan, 2-stage software pipeline ----
    // wave w owns points [w*STRIP, (w+1)*STRIP), appends masked indices in
    // original order into its private LDS segment.
    const size_t rowbase = (size_t)b * G_;
    int* wseg = s_list + wav * STRIP;
    int wbase = 0;                              // uniform within wave
    const int laneoff = (lane < 16) ? 0 : 2;    // {x,y} or {z,pad}
    const float* lptr = gp + (rowbase + (size_t)(wav * STRIP + ncol)) * 6 + laneoff;

    float2 cur = *reinterpret_cast<const float2*>(lptr);   // chunk 0 in flight
    for (int chunk = 0; chunk < NCHUNK; ++chunk) {
        // issue next chunk's load before consuming the current one
        const int nextc = (chunk + 1 < NCHUNK) ? chunk + 1 : chunk;
        const float2 nxt = *reinterpret_cast<const float2*>(
            lptr + (size_t)nextc * PTS * 6);

        const int g0 = wav * STRIP + chunk * PTS;
        v2f amat;
        amat.x = cur.x;
        amat.y = (lane < 16) ? cur.y : 0.f;     // K=3 pad column

        v8f dmat = __builtin_amdgcn_wmma_f32_16x16x4_f32(
            false, amat, false, bmat, (short)0, cmat, false, false);

        // point M = r (+8 hi half): x,y,z live on lanes {0,1,2}/{16,17,18} of dmat[r]
        unsigned m16 = 0u;
#pragma unroll
        for (int r = 0; r < 8; ++r) {
            float val = dmat[r];
            unsigned bal = __builtin_amdgcn_ballot_w32(val > blo)
                         & __builtin_amdgcn_ballot_w32(val < bhi);
            unsigned t = bal & (bal >> 1) & (bal >> 2);   // AND of 3 coord tests
            m16 |= (t & 1u) << r;
            m16 |= ((t >> 16) & 1u) << (r + 8);
        }
        if (lane < 16 && ((m16 >> lane) & 1u)) {
            int pos = wbase + __builtin_popcount(m16 & ((1u << lane) - 1u));
            wseg[pos] = g0 + lane;
        }
        wbase += __builtin_popcount(m16);
        cur = nxt;
    }
    if (lane == 0) s_cnt[wav] = wbase;
    __syncthreads();

    // ---- merge view: exclusive prefix over the 8 wave segments ----
    int pref[WAVES + 1];
    pref[0] = 0;
#pragma unroll
    for (int w = 0; w < WAVES; ++w) pref[w + 1] = pref[w] + s_cnt[w];
    const int  count = pref[WAVES];
    const bool valid = count > 5;

    // ---- output phase (gathers hit L2: group_points ~100MB < 192MB L2) ----
    const float m00 = s_mat[0], m01 = s_mat[1], m02 = s_mat[2];
    const float m10 = s_mat[3], m11 = s_mat[4], m12 = s_mat[5];
    const float m20 = s_mat[6], m21 = s_mat[7], m22 = s_mat[8];
    const float ccx = s_mat[9], ccy = s_mat[10], ccz = s_mat[11];

    float* pc   = out_pc    + (size_t)b * REGION * 6;
    float* oidx = out_idx   + (size_t)b * REGION;
    float* oin  = out_inall + (size_t)b * REGION;

    for (int j = tid; j < REGION; j += THREADS) {
        if (valid) {
            int jm = (int)((unsigned)j % (unsigned)count);
            int w = 0;
#pragma unroll
            for (int k = 0; k < WAVES - 1; ++k) if (jm >= pref[k + 1]) w = k + 1;
            int g = s_list[w * STRIP + (jm - pref[w])];
            const float* p = gp + (rowbase + (size_t)g) * 6;
            float px = p[0] - ccx, py = p[1] - ccy, pz = p[2] - ccz;
            float tx = fmaf(m00, px, fmaf(m01, py, m02 * pz));
            float ty = fmaf(m10, px, fmaf(m11, py, m12 * pz));
            float tz = fmaf(m20, px, fmaf(m21, py, m22 * pz));
            pc[j*6 + 0] = tx;
            pc[j*6 + 1] = ty;
            pc[j*6 + 2] = tz;
            pc[j*6 + 3] = p[3];
            pc[j*6 + 4] = p[4];
            pc[j*6 + 5] = p[5];
            oidx[j] = (float)g;
            oin[j]  = (float)gindex[rowbase + (size_t)g];
        } else {
#pragma unroll
            for (int k = 0; k < 6; ++k) pc[j*6 + k] = -1.0f;
            oidx[j] = -1.0f;
            oin[j]  = -1.0f;
        }
    }
    if (tid == 0) out_mask[b] = valid ? 1.0f : 0.0f;
}

extern "C" void kernel_launch(void* const* d_in, const int* in_sizes, int n_in,
                              void* d_out, int out_size, void* d_ws, size_t ws_size,
                              hipStream_t stream) {
    (void)in_sizes; (void)n_in; (void)d_ws; (void)ws_size; (void)out_size;
    const float* gp     = (const float*)d_in[0];
    const int*   gindex = (const int*)  d_in[1];
    const float* grasp  = (const float*)d_in[2];
    const float* widths = (const float*)d_in[3];
    const float* height = (const float*)d_in[4];
    const float* depths = (const float*)d_in[5];

    float* out       = (float*)d_out;
    float* out_pc    = out;                                    // B*REGION*6
    float* out_idx   = out_pc  + (size_t)B_ * REGION * 6;      // B*REGION
    float* out_inall = out_idx + (size_t)B_ * REGION;          // B*REGION
    float* out_mask  = out_inall + (size_t)B_ * REGION;        // B

    hipLaunchKernelGGL(gripper_region_kernel, dim3(B_), dim3(THREADS), 0, stream,
                       gp, gindex, grasp, widths, height, depths,
                       out_pc, out_idx, out_inall, out_mask);
}